// BrainGCN_81913616269326
// MI455X (gfx1250) — compile-verified
//
#include <hip/hip_runtime.h>
#include <hip/hip_bf16.h>
#include <math.h>

// Problem constants (match reference file)
#define NN 100000
#define NE 1600000

typedef __attribute__((ext_vector_type(2))) float v2f;
typedef __attribute__((ext_vector_type(8))) float v8f;

// ---------------- small utility kernels ----------------

__global__ void zero_f(float* __restrict__ p, int n) {
    int i = blockIdx.x * blockDim.x + threadIdx.x;
    if (i < n) p[i] = 0.0f;
}

__global__ void deg_count(const int* __restrict__ dst, float* __restrict__ deg, int e_cnt) {
    int e = blockIdx.x * blockDim.x + threadIdx.x;
    if (e < e_cnt) atomicAdd(&deg[dst[e]], 1.0f);
}

// dinv = rsqrt(deg+1) (edge norm), selfn = 1/(deg+1) (self-loop norm)
__global__ void deg_norm(const float* __restrict__ deg, float* __restrict__ dinv,
                         float* __restrict__ selfn, int n) {
    int i = blockIdx.x * blockDim.x + threadIdx.x;
    if (i >= n) return;
    float d = deg[i] + 1.0f;
    dinv[i]  = rsqrtf(d);
    selfn[i] = 1.0f / d;
}

// ---------------- f32 WMMA GEMM: Y = act(X[MxK] @ W[KxNout] + bias) ----------------
// One wave (32 lanes) owns one 16x16 output tile, accumulating with
// V_WMMA_F32_16X16X4_F32 over K in steps of 4. Full f32 precision.
// A layout (16x4 f32): lanes 0-15 -> M=lane, VGPR0=K+0, VGPR1=K+1
//                      lanes 16-31 -> M=lane-16, VGPR0=K+2, VGPR1=K+3
// B layout (4x16 f32): lanes 0-15 -> N=lane, VGPR0=K+0 row, VGPR1=K+1 row
//                      lanes 16-31 -> N=lane-16, VGPR0=K+2, VGPR1=K+3
// C/D layout: VGPR r, lanes 0-15 -> row m0+r; lanes 16-31 -> row m0+r+8.
__global__ void wmma_gemm_f32(const float* __restrict__ X, const float* __restrict__ W,
                              const float* __restrict__ bias, float* __restrict__ Y,
                              int M, int K, int Nout, int act) {
    int wave  = (int)((blockIdx.x * blockDim.x + threadIdx.x) >> 5);
    int lane  = threadIdx.x & 31;
    int tilesN = Nout >> 4;
    int m0 = (wave / tilesN) << 4;
    int n0 = (wave % tilesN) << 4;
    if (m0 >= M) return;              // wave-uniform: EXEC stays all-1s for WMMA
    int half = lane >> 4;             // which half-wave
    int l    = lane & 15;

    const float* xrow = X + (long long)(m0 + l) * K;
    v8f c = {};
    for (int k = 0; k < K; k += 4) {
        int ka = k + half * 2;
        v2f a, b;
        a.x = xrow[ka];
        a.y = xrow[ka + 1];
        b.x = W[ka * Nout + n0 + l];
        b.y = W[(ka + 1) * Nout + n0 + l];
        // 8 args: (neg_a, A, neg_b, B, c_mod, C, reuse_a, reuse_b)
        c = __builtin_amdgcn_wmma_f32_16x16x4_f32(false, a, false, b,
                                                  (short)0, c, false, false);
    }
    int col = n0 + l;
    float bv = bias ? bias[col] : 0.0f;
#pragma unroll
    for (int r = 0; r < 8; ++r) {
        int row = m0 + r + half * 8;
        float v = c[r] + bv;
        if (act) v = tanhf(v);
        Y[(long long)row * Nout + col] = v;
    }
}

// ---------------- edge aggregation (the bandwidth-dominant step) ----------------
// 64 consecutive threads handle one edge: coalesced 256B gather from xl[src],
// coalesced atomic-add burst into agg[dst]. F fixed at 64.
__global__ void scatter_agg64(const float* __restrict__ xl, const int* __restrict__ src,
                              const int* __restrict__ dst, const float* __restrict__ dinv,
                              float* __restrict__ agg, int e_cnt) {
    int t = blockIdx.x * blockDim.x + threadIdx.x;
    int e = t >> 6;
    if (e >= e_cnt) return;
    int f = t & 63;
    int s = src[e];
    int d = dst[e];
    float norm = dinv[s] * dinv[d];
    atomicAdd(&agg[(long long)d * 64 + f], xl[(long long)s * 64 + f] * norm);
}

// out = tanh(agg + xl * selfn[node] + bias[f]), over N*64 elements
__global__ void conv_finalize64(const float* __restrict__ agg, const float* __restrict__ xl,
                                const float* __restrict__ selfn, const float* __restrict__ bias,
                                float* __restrict__ out, int n_elems) {
    int t = blockIdx.x * blockDim.x + threadIdx.x;
    if (t >= n_elems) return;
    int node = t >> 6;
    int f    = t & 63;
    out[t] = tanhf(agg[t] + xl[t] * selfn[node] + bias[f]);
}

// final FC: out[n] = h[n,0:32] . w + b   (Nout=1, cheap VALU dot)
__global__ void fc_last32(const float* __restrict__ h, const float* __restrict__ w,
                          const float* __restrict__ b, float* __restrict__ out, int n) {
    int i = blockIdx.x * blockDim.x + threadIdx.x;
    if (i >= n) return;
    float acc = b[0];
#pragma unroll
    for (int j = 0; j < 32; ++j) acc += h[i * 32 + j] * w[j];
    out[i] = acc;
}

// ---------------- launcher ----------------

extern "C" void kernel_launch(void* const* d_in, const int* in_sizes, int n_in,
                              void* d_out, int out_size, void* d_ws, size_t ws_size,
                              hipStream_t stream) {
    const float* x   = (const float*)d_in[0];
    const int*   ei  = (const int*)d_in[1];   // [2, E] flat: src then dst
    const float* cW0 = (const float*)d_in[2];
    const float* cb0 = (const float*)d_in[3];
    const float* cW1 = (const float*)d_in[4];
    const float* cb1 = (const float*)d_in[5];
    const float* fW0 = (const float*)d_in[6];
    const float* fb0 = (const float*)d_in[7];
    const float* fW1 = (const float*)d_in[8];
    const float* fb1 = (const float*)d_in[9];
    const float* fW2 = (const float*)d_in[10];
    const float* fb2 = (const float*)d_in[11];
    const int* src = ei;
    const int* dst = ei + NE;

    // workspace layout (floats)
    float* ws    = (float*)d_ws;
    float* xl    = ws;                          // N*64  linear-transform output
    float* agg   = xl  + (size_t)NN * 64;       // N*64  neighbor aggregation
    float* h1    = agg + (size_t)NN * 64;       // N*64
    float* h2    = h1  + (size_t)NN * 64;       // N*64
    float* deg   = h2  + (size_t)NN * 64;       // N
    float* dinv  = deg  + NN;                   // N
    float* selfn = dinv + NN;                   // N

    const int B = 256;
    dim3 blk(B);
    auto cdiv = [](long long a, long long b) { return (unsigned)((a + b - 1) / b); };

    unsigned gN    = cdiv(NN, B);
    unsigned gE    = cdiv(NE, B);
    unsigned gNF   = cdiv((long long)NN * 64, B);
    unsigned gEF   = cdiv((long long)NE * 64, B);
    unsigned gG64  = cdiv((long long)(NN / 16) * 4 * 32, B);  // Nout=64 tiles
    unsigned gG32  = cdiv((long long)(NN / 16) * 2 * 32, B);  // Nout=32 tiles

    // degrees + norms (shared by both conv layers)
    zero_f<<<gN, blk, 0, stream>>>(deg, NN);
    deg_count<<<gE, blk, 0, stream>>>(dst, deg, NE);
    deg_norm<<<gN, blk, 0, stream>>>(deg, dinv, selfn, NN);

    // ---- conv layer 0: h1 = tanh(agg(x@cW0) + self + cb0) ----
    wmma_gemm_f32<<<gG64, blk, 0, stream>>>(x, cW0, nullptr, xl, NN, 128, 64, 0);
    zero_f<<<gNF, blk, 0, stream>>>(agg, NN * 64);
    scatter_agg64<<<gEF, blk, 0, stream>>>(xl, src, dst, dinv, agg, NE);
    conv_finalize64<<<gNF, blk, 0, stream>>>(agg, xl, selfn, cb0, h1, NN * 64);

    // ---- conv layer 1: h2 = tanh(agg(h1@cW1) + self + cb1) ----
    wmma_gemm_f32<<<gG64, blk, 0, stream>>>(h1, cW1, nullptr, xl, NN, 64, 64, 0);
    zero_f<<<gNF, blk, 0, stream>>>(agg, NN * 64);
    scatter_agg64<<<gEF, blk, 0, stream>>>(xl, src, dst, dinv, agg, NE);
    conv_finalize64<<<gNF, blk, 0, stream>>>(agg, xl, selfn, cb1, h2, NN * 64);

    // ---- MLP head ----
    wmma_gemm_f32<<<gG64, blk, 0, stream>>>(h2, fW0, fb0, h1, NN, 64, 64, 1); // tanh
    wmma_gemm_f32<<<gG32, blk, 0, stream>>>(h1, fW1, fb1, xl, NN, 64, 32, 1); // tanh
    fc_last32<<<gN, blk, 0, stream>>>(xl, fW2, fb2, (float*)d_out, NN);
}